// Net_88665304859397
// MI455X (gfx1250) — compile-verified
//
#include <hip/hip_runtime.h>
#include <stdint.h>

#define N_ATOMS 50000
#define N_EDGES 800000
#define NB 128
#define NF 128
#define NG 32
#define NC 3

typedef __attribute__((ext_vector_type(2))) float v2f;
typedef __attribute__((ext_vector_type(8))) float v8f;

__device__ __forceinline__ float sspf(float x) {
    // softplus(x) - ln(2), numerically stable
    float sp = fmaxf(x, 0.0f) + log1pf(__expf(-fabsf(x)));
    return sp - 0.69314718055994531f;
}

// Multi-M-tile wave GEMM via V_WMMA_F32_16X16X4_F32.
// As: LDS row-major [MT*16][LDA] (A tiles), B: global row-major [K][LDB].
// Each B fragment is reused across MT accumulators (register blocking),
// cutting vmem issue rate and weight re-reads by MT and avoiding D->D stalls.
// A layout : lane = 16*K[1] + M, vgpr j = K[0]       -> a[j] = A[rw][2*hl + j]
// B layout : vgpr j, lanes0-15 row j, lanes16-31 j+2 -> b[j] = B[k + 2*hl + j][n]
template<int K, int LDA, int LDB, int MT>
__device__ __forceinline__ void wmma_multi(const float* __restrict__ As,
                                           const float* __restrict__ B,
                                           int n0, v8f* acc) {
    const int lane = threadIdx.x & 31;
    const int hl   = lane >> 4;
    const int rw   = lane & 15;
    #pragma unroll
    for (int k = 0; k < K; k += 4) {
        v2f b;
        b.x = B[(size_t)(k + 2 * hl + 0) * LDB + n0 + rw];
        b.y = B[(size_t)(k + 2 * hl + 1) * LDB + n0 + rw];
        #pragma unroll
        for (int mt = 0; mt < MT; ++mt) {
            v2f a;
            a.x = As[(mt * 16 + rw) * LDA + k + 2 * hl + 0];
            a.y = As[(mt * 16 + rw) * LDA + k + 2 * hl + 1];
            acc[mt] = __builtin_amdgcn_wmma_f32_16x16x4_f32(false, a, false, b,
                                                            (short)0, acc[mt],
                                                            false, false);
        }
    }
}

// Stage a contiguous 16x128 f32 tile (row-major) into LDS (stride 132) using
// gfx1250 async copy: GLOBAL_LOAD_ASYNC_TO_LDS_B128 (ASYNCcnt), then wait.
// A __shared__ pointer's low 32 bits are the LDS byte offset.
template<int THREADS>
__device__ __forceinline__ void stage_tile_async(float* dst,
                                                 const float* __restrict__ src) {
    #pragma unroll
    for (int it = 0; it < (512 / THREADS); ++it) {
        int chunk = it * THREADS + (int)threadIdx.x;   // 512 x 16B chunks
        int row = chunk >> 5;                          // 32 chunks per 128-f row
        int c4  = chunk & 31;
        uint32_t lds_off = (uint32_t)(uintptr_t)(dst + row * 132 + c4 * 4);
        const float* gp = src + (size_t)row * 128 + c4 * 4;
        asm volatile("global_load_async_to_lds_b128 %0, %1, off"
                     :: "v"(lds_off), "v"(gp)
                     : "memory");
    }
    asm volatile("s_wait_asynccnt 0x0" ::: "memory");
}

// h = embed[r]  (grid: N_ATOMS, block: NB)
__global__ void k_embed(const int* __restrict__ r, const float* __restrict__ embed,
                        float* __restrict__ h) {
    int atom = blockIdx.x;
    h[(size_t)atom * NB + threadIdx.x] = embed[(size_t)r[atom] * NB + threadIdx.x];
}

// d[e] = ||xyz[a0]-xyz[a1]|| with EPS
__global__ void k_dist(const float* __restrict__ xyz, const int* __restrict__ a,
                       float* __restrict__ d) {
    int e = blockIdx.x * blockDim.x + threadIdx.x;
    if (e < N_EDGES) {
        int i0 = a[2 * e + 0], i1 = a[2 * e + 1];
        float dx = xyz[3 * i0 + 0] - xyz[3 * i1 + 0];
        float dy = xyz[3 * i0 + 1] - xyz[3 * i1 + 1];
        float dz = xyz[3 * i0 + 2] - xyz[3 * i1 + 2];
        d[e] = sqrtf(dx * dx + dy * dy + dz * dz + 1e-12f);
    }
}

__global__ void k_zero(float* __restrict__ p, int n) {
    int i = blockIdx.x * blockDim.x + threadIdx.x;
    if (i < n) p[i] = 0.0f;
}

// hf = h @ afw + afb   (grid: N_ATOMS/16, block: 256 = 8 waves covering N=128)
__global__ void __launch_bounds__(256) k_hf(const float* __restrict__ h,
                                            const float* __restrict__ afw,
                                            const float* __restrict__ afb,
                                            float* __restrict__ hf) {
    __shared__ float As[16 * 132];
    const int m0 = blockIdx.x * 16;
    stage_tile_async<256>(As, h + (size_t)m0 * NB);
    __syncthreads();
    const int n0 = (threadIdx.x >> 5) * 16;
    v8f acc[1] = {};
    wmma_multi<128, 132, 128, 1>(As, afw, n0, acc);
    const int lane = threadIdx.x & 31, hl = lane >> 4, rw = lane & 15;
    float bias = afb[n0 + rw];
    #pragma unroll
    for (int p = 0; p < 8; ++p)
        hf[(size_t)(m0 + p + 8 * hl) * NB + n0 + rw] = acc[0][p] + bias;
}

// Fused edge pipeline, 64 edges per workgroup (4 M-tiles register-blocked):
//   g -> t1 = ssp(g@fw1+fb1) -> W = t1@fw2+fb2 -> msg = hf[src]*W -> atomicAdd agg[dst]
__global__ void __launch_bounds__(256) k_edge(const float* __restrict__ d,
                                              const int* __restrict__ a,
                                              const float* __restrict__ fw1,
                                              const float* __restrict__ fb1,
                                              const float* __restrict__ fw2,
                                              const float* __restrict__ fb2,
                                              const float* __restrict__ hf,
                                              float* __restrict__ agg) {
    __shared__ float gs[64 * 36];
    __shared__ float t1[64 * 132];
    const int e0 = blockIdx.x * 64;
    const float width = 5.0f / 31.0f;               // linspace(0, CUTOFF, NG) step
    const float coeff = -0.5f / (width * width);
    for (int idx = threadIdx.x; idx < 64 * NG; idx += 256) {
        int row = idx >> 5, col = idx & 31;
        float t = d[e0 + row] - (float)col * width;
        gs[row * 36 + col] = __expf(coeff * t * t);
    }
    __syncthreads();
    const int n0 = (threadIdx.x >> 5) * 16;
    const int lane = threadIdx.x & 31, hl = lane >> 4, rw = lane & 15;

    v8f acc[4] = {};
    wmma_multi<32, 36, 128, 4>(gs, fw1, n0, acc);
    float b1 = fb1[n0 + rw];
    #pragma unroll
    for (int mt = 0; mt < 4; ++mt)
        #pragma unroll
        for (int p = 0; p < 8; ++p)
            t1[(mt * 16 + p + 8 * hl) * 132 + n0 + rw] = sspf(acc[mt][p] + b1);
    __syncthreads();

    v8f acc2[4] = {};
    wmma_multi<128, 132, 128, 4>(t1, fw2, n0, acc2);
    float b2 = fb2[n0 + rw];
    #pragma unroll
    for (int mt = 0; mt < 4; ++mt) {
        #pragma unroll
        for (int p = 0; p < 8; ++p) {
            int e   = e0 + mt * 16 + p + 8 * hl;
            int src = a[2 * e + 1];
            int dst = a[2 * e + 0];
            float msg = hf[(size_t)src * NB + n0 + rw] * (acc2[mt][p] + b2);
            atomicAdd(&agg[(size_t)dst * NB + n0 + rw], msg);
        }
    }
}

// h += ssp(agg@ow1+ob1)@ow2+ob2   (grid: N_ATOMS/16, block: 256)
__global__ void __launch_bounds__(256) k_atom(const float* __restrict__ agg,
                                              const float* __restrict__ ow1,
                                              const float* __restrict__ ob1,
                                              const float* __restrict__ ow2,
                                              const float* __restrict__ ob2,
                                              float* __restrict__ h) {
    __shared__ float As[16 * 132];
    __shared__ float t1[16 * 132];
    const int m0 = blockIdx.x * 16;
    stage_tile_async<256>(As, agg + (size_t)m0 * NB);
    __syncthreads();
    const int n0 = (threadIdx.x >> 5) * 16;
    const int lane = threadIdx.x & 31, hl = lane >> 4, rw = lane & 15;
    v8f acc[1] = {};
    wmma_multi<128, 132, 128, 1>(As, ow1, n0, acc);
    float b1 = ob1[n0 + rw];
    #pragma unroll
    for (int p = 0; p < 8; ++p)
        t1[(p + 8 * hl) * 132 + n0 + rw] = sspf(acc[0][p] + b1);
    __syncthreads();
    v8f acc2[1] = {};
    wmma_multi<128, 132, 128, 1>(t1, ow2, n0, acc2);
    float b2 = ob2[n0 + rw];
    #pragma unroll
    for (int p = 0; p < 8; ++p)
        h[(size_t)(m0 + p + 8 * hl) * NB + n0 + rw] += acc2[0][p] + b2;
}

// y = ssp(h@aw1+ab1)@aw2+ab2; block-sum via atomicAdd into out[atom / n_per]
// grid: N_ATOMS/16, block: 128 (4 waves covering N=64)
__global__ void __launch_bounds__(128) k_out(const float* __restrict__ h,
                                             const float* __restrict__ aw1,
                                             const float* __restrict__ ab1,
                                             const float* __restrict__ aw2,
                                             const float* __restrict__ ab2,
                                             const int* __restrict__ n_per_p,
                                             float* __restrict__ out) {
    __shared__ float As[16 * 132];
    __shared__ float t1[16 * 68];
    const int m0 = blockIdx.x * 16;
    stage_tile_async<128>(As, h + (size_t)m0 * NB);
    __syncthreads();
    const int n0 = (threadIdx.x >> 5) * 16;  // 4 waves cover 64 cols
    const int lane = threadIdx.x & 31, hl = lane >> 4, rw = lane & 15;
    v8f acc[1] = {};
    wmma_multi<128, 132, 64, 1>(As, aw1, n0, acc);
    float b1 = ab1[n0 + rw];
    #pragma unroll
    for (int p = 0; p < 8; ++p)
        t1[(p + 8 * hl) * 68 + n0 + rw] = sspf(acc[0][p] + b1);
    __syncthreads();
    if (threadIdx.x < 16) {
        float y = ab2[0];
        #pragma unroll 8
        for (int j = 0; j < 64; ++j)
            y += t1[threadIdx.x * 68 + j] * aw2[j];
        int npp = n_per_p[0];
        atomicAdd(&out[(m0 + threadIdx.x) / npp], y);
    }
}

extern "C" void kernel_launch(void* const* d_in, const int* in_sizes, int n_in,
                              void* d_out, int out_size, void* d_ws, size_t ws_size,
                              hipStream_t stream) {
    const int*   r     = (const int*)  d_in[0];
    const float* xyz   = (const float*)d_in[1];
    const int*   a     = (const int*)  d_in[2];
    const int*   nper  = (const int*)  d_in[3];
    const float* embed = (const float*)d_in[4];
    const float* fw1   = (const float*)d_in[5];
    const float* fb1   = (const float*)d_in[6];
    const float* fw2   = (const float*)d_in[7];
    const float* fb2   = (const float*)d_in[8];
    const float* afw   = (const float*)d_in[9];
    const float* afb   = (const float*)d_in[10];
    const float* ow1   = (const float*)d_in[11];
    const float* ob1   = (const float*)d_in[12];
    const float* ow2   = (const float*)d_in[13];
    const float* ob2   = (const float*)d_in[14];
    const float* aw1   = (const float*)d_in[15];
    const float* ab1   = (const float*)d_in[16];
    const float* aw2   = (const float*)d_in[17];
    const float* ab2   = (const float*)d_in[18];
    float* out = (float*)d_out;

    // workspace layout (floats): d | h | hf | agg  (~80 MB)
    float* ws  = (float*)d_ws;
    float* dd  = ws;
    float* h   = dd + N_EDGES;
    float* hf  = h + (size_t)N_ATOMS * NB;
    float* agg = hf + (size_t)N_ATOMS * NB;

    k_embed<<<N_ATOMS, NB, 0, stream>>>(r, embed, h);
    k_dist<<<(N_EDGES + 255) / 256, 256, 0, stream>>>(xyz, a, dd);

    for (int i = 0; i < NC; ++i) {
        k_hf<<<N_ATOMS / 16, 256, 0, stream>>>(
            h, afw + (size_t)i * NB * NF, afb + (size_t)i * NF, hf);
        k_zero<<<(N_ATOMS * NB + 255) / 256, 256, 0, stream>>>(agg, N_ATOMS * NB);
        k_edge<<<N_EDGES / 64, 256, 0, stream>>>(
            dd, a,
            fw1 + (size_t)i * NG * NF, fb1 + (size_t)i * NF,
            fw2 + (size_t)i * NF * NF, fb2 + (size_t)i * NF,
            hf, agg);
        k_atom<<<N_ATOMS / 16, 256, 0, stream>>>(
            agg,
            ow1 + (size_t)i * NF * NB, ob1 + (size_t)i * NB,
            ow2 + (size_t)i * NB * NB, ob2 + (size_t)i * NB,
            h);
    }

    k_zero<<<1, 32, 0, stream>>>(out, out_size);
    k_out<<<N_ATOMS / 16, 128, 0, stream>>>(h, aw1, ab1, aw2, ab2, nper, out);
}